// GCNEncoderSimple_58248346469020
// MI455X (gfx1250) — compile-verified
//
#include <hip/hip_runtime.h>
#include <hip/hip_bf16.h>

typedef __attribute__((ext_vector_type(2))) float v2f;
typedef __attribute__((ext_vector_type(8))) float v8f;

#define N_NODES 50000
#define N_EDGES 800000
#define N_FEAT  512
#define N_HID   256
#define N_LAT   128

// ---------------------------------------------------------------------------
// Zero-fill (graph-capture safe; we never rely on harness poison state)
// ---------------------------------------------------------------------------
__global__ void fill_zero_kernel(float* __restrict__ p, long long n) {
    long long i = (long long)blockIdx.x * blockDim.x + threadIdx.x;
    if (i < n) p[i] = 0.0f;
}

// ---------------------------------------------------------------------------
// Dense GEMM C[M,N] = A[M,K] * B[K,N] using V_WMMA_F32_16X16X4_F32.
//
// Each wave computes a 16x64 output strip = 4 accumulator tiles (32 VGPRs),
// so one A fragment load feeds 4 WMMAs (4x less A traffic than 1 tile/wave,
// WMMA:VMEM ratio 4:9 per K-step). Block = (32,5) waves; grid.x covers M
// tiles (3125 = 625*5 exactly), grid.y covers N strips of 64. All tilings
// are exact so EXEC is all-ones around every WMMA (ISA requirement).
//
// Fragment layouts (CDNA5 ISA 7.12.2, 32-bit data):
//   A 16x4 (MxK):  lanes 0-15 row M=lane,    VGPR0=K0 VGPR1=K1;
//                  lanes 16-31 row M=lane-16, VGPR0=K2 VGPR1=K3.
//   B 4x16 (KxN):  lanes 0-15 col N=lane,    VGPR0=row K0, VGPR1=row K1;
//                  lanes 16-31 col N=lane-16, VGPR0=K2,    VGPR1=K3.
//   C 16x16: VGPR r -> row (r + 8*laneHi), col = lane&15.
//
// FUSE_RELU_BIAS: A element becomes relu(A[row][k] + bias[k]) (bias indexed
// by K), letting layer-2 consume layer-1's pre-activation accumulator
// without an extra 51MB elementwise pass.
// ---------------------------------------------------------------------------
template <int K, int N, bool FUSE_RELU_BIAS>
__global__ void gemm_wmma_f32_strip_kernel(const float* __restrict__ A,
                                           const float* __restrict__ B,
                                           const float* __restrict__ bias,
                                           float* __restrict__ C) {
    const int lane   = threadIdx.x;              // 0..31
    const int mt     = blockIdx.x * 5 + threadIdx.y;   // M tile index
    const int laneLo = lane & 15;
    const int laneHi = lane >> 4;                // 0 or 1

    const long long arow = (long long)(mt * 16 + laneLo);
    const float* __restrict__ aRow = A + arow * K;
    // base column of this wave's 64-wide strip
    const int bcol = blockIdx.y * 64 + laneLo;

    v8f acc[4] = {{}, {}, {}, {}};

#pragma unroll 4
    for (int k = 0; k < K; k += 4) {
        const int ka = k + laneHi * 2;           // this lane-half's K pair
        // A fragment: contiguous pair along K, shared by all 4 tiles
        v2f a = *(const v2f*)(aRow + ka);
        if (FUSE_RELU_BIAS) {
            float t0 = a.x + bias[ka];
            float t1 = a.y + bias[ka + 1];
            a.x = t0 > 0.0f ? t0 : 0.0f;
            a.y = t1 > 0.0f ? t1 : 0.0f;
        }
        const float* __restrict__ bRow0 = B + (long long)ka * N + bcol;
        const float* __restrict__ bRow1 = bRow0 + N;
#pragma unroll
        for (int t = 0; t < 4; ++t) {
            v2f b;
            b.x = bRow0[t * 16];
            b.y = bRow1[t * 16];
            acc[t] = __builtin_amdgcn_wmma_f32_16x16x4_f32(
                /*neg_a=*/false, a, /*neg_b=*/false, b,
                /*c_mod=*/(short)0, acc[t],
                /*reuse_a=*/false, /*reuse_b=*/false);
        }
    }

#pragma unroll
    for (int t = 0; t < 4; ++t) {
#pragma unroll
        for (int r = 0; r < 8; ++r) {
            const long long crow = (long long)(mt * 16 + r + laneHi * 8);
            C[crow * N + bcol + t * 16] = acc[t][r];
        }
    }
}

// ---------------------------------------------------------------------------
// COO SpMM scatter: out[rows[e], :] += vals[e] * dense[cols[e], :]
// One wave per edge; lane j handles features j, j+32, ... (fully coalesced
// loads and global_atomic_add_f32 scatter; accumulation lives in the 192MB
// L2 -- the [50000,D] dense operand is 25-51MB, fully L2 resident).
// ---------------------------------------------------------------------------
template <int D>
__global__ void spmm_atomic_kernel(const int* __restrict__ rows,
                                   const int* __restrict__ cols,
                                   const float* __restrict__ vals,
                                   const float* __restrict__ dense,
                                   float* __restrict__ out, int nEdges) {
    const int e = blockIdx.x * 8 + threadIdx.y;
    if (e >= nEdges) return;
    const int lane = threadIdx.x;
    const int r = rows[e];
    const int c = cols[e];
    const float v = vals[e];
    const float* __restrict__ src = dense + (long long)c * D;
    float* __restrict__ dst = out + (long long)r * D;
#pragma unroll
    for (int j = 0; j < D / 32; ++j) {
        const int f = lane + j * 32;
        atomicAdd(&dst[f], v * src[f]);
    }
}

// ---------------------------------------------------------------------------
// Final elementwise: p[i] = relu(p[i] + b[i & (D-1)])   (D power of two)
// ---------------------------------------------------------------------------
template <int D>
__global__ void bias_relu_kernel(float* __restrict__ p,
                                 const float* __restrict__ b, long long n) {
    long long i = (long long)blockIdx.x * blockDim.x + threadIdx.x;
    if (i < n) {
        float t = p[i] + b[i & (D - 1)];
        p[i] = t > 0.0f ? t : 0.0f;
    }
}

// ---------------------------------------------------------------------------
// Launch: x, edge_rows, edge_cols, edge_vals, W1, b1, W2, b2 -> z [50000,128]
// ---------------------------------------------------------------------------
extern "C" void kernel_launch(void* const* d_in, const int* in_sizes, int n_in,
                              void* d_out, int out_size, void* d_ws, size_t ws_size,
                              hipStream_t stream) {
    const float* x  = (const float*)d_in[0];
    const int*   er = (const int*)d_in[1];
    const int*   ec = (const int*)d_in[2];
    const float* ev = (const float*)d_in[3];
    const float* W1 = (const float*)d_in[4];
    const float* b1 = (const float*)d_in[5];
    const float* W2 = (const float*)d_in[6];
    const float* b2 = (const float*)d_in[7];
    float* out = (float*)d_out;

    // Workspace layout: [xw1 | h], hw2 reuses xw1's slot after SpMM1.
    float* xw1 = (float*)d_ws;                                  // 50000*256
    float* h   = xw1 + (long long)N_NODES * N_HID;              // 50000*256
    float* hw2 = xw1;                                           // 50000*128

    const dim3 gemmBlk(32, 5);   // 5 waves, each owning one 16x64 strip
    const dim3 spmmBlk(32, 8);   // 8 waves, each owning one edge

    // 1) h = 0
    {
        long long n = (long long)N_NODES * N_HID;
        fill_zero_kernel<<<(unsigned)((n + 255) / 256), 256, 0, stream>>>(h, n);
    }
    // 2) xw1 = x @ W1        grid: 625 blocks * 5 waves = 3125 M tiles; 4 strips
    gemm_wmma_f32_strip_kernel<N_FEAT, N_HID, false>
        <<<dim3(N_NODES / 16 / 5, N_HID / 64), gemmBlk, 0, stream>>>(x, W1, nullptr, xw1);
    // 3) h += scatter(edges, xw1)
    spmm_atomic_kernel<N_HID>
        <<<(N_EDGES + 7) / 8, spmmBlk, 0, stream>>>(er, ec, ev, xw1, h, N_EDGES);
    // 4) out = 0
    {
        long long n = (long long)N_NODES * N_LAT;
        fill_zero_kernel<<<(unsigned)((n + 255) / 256), 256, 0, stream>>>(out, n);
    }
    // 5) hw2 = relu(h + b1) @ W2      grid: 625 x 2 strips
    gemm_wmma_f32_strip_kernel<N_HID, N_LAT, true>
        <<<dim3(N_NODES / 16 / 5, N_LAT / 64), gemmBlk, 0, stream>>>(h, W2, b1, hw2);
    // 6) out += scatter(edges, hw2)
    spmm_atomic_kernel<N_LAT>
        <<<(N_EDGES + 7) / 8, spmmBlk, 0, stream>>>(er, ec, ev, hw2, out, N_EDGES);
    // 7) out = relu(out + b2)
    {
        long long n = (long long)N_NODES * N_LAT;
        bias_relu_kernel<N_LAT>
            <<<(unsigned)((n + 255) / 256), 256, 0, stream>>>(out, b2, n);
    }
}